// MinibatchDiscrimination_55972013801825
// MI455X (gfx1250) — compile-verified
//
#include <hip/hip_runtime.h>

typedef __attribute__((ext_vector_type(2))) float v2f;
typedef __attribute__((ext_vector_type(8))) float v8f;

#define N_BATCH 2048
#define A_DIM   512
#define B_DIM   64
#define C_DIM   16
#define N_COLS  (B_DIM * C_DIM)   // 1024

// ---------------------------------------------------------------------------
// Phase 1: M2[b][i][c] = sum_k x[i][k] * W[k][b*16+c]   (W = T viewed [512][1024])
// One wave computes one 16x16 output tile with V_WMMA_F32_16X16X4_F32,
// looping K in steps of 4. Block = 8 waves covering 8 adjacent N-tiles that
// share the same 16-row X tile, which is staged through LDS (32 KB).
// Since C_DIM == 16, N-tile index tj IS the b index and col IS c, so the
// store lands directly in the b-major layout phase 2 wants.
// ---------------------------------------------------------------------------
__global__ __launch_bounds__(256)
void mbd_gemm_wmma(const float* __restrict__ x,
                   const float* __restrict__ W,
                   float* __restrict__ M2) {
    __shared__ float sX[16 * A_DIM];            // 32 KB: 16 rows of x

    const int tid = threadIdx.x;
    const int ti  = blockIdx.x >> 3;            // 0..127  (i tile)
    const int tjg = blockIdx.x & 7;             // group of 8 N-tiles

    // Cooperative load of x rows [ti*16, ti*16+16) : 8192 floats = 2048 float4
    const float4* xs = (const float4*)(x + (size_t)ti * 16 * A_DIM);
    float4*       xd = (float4*)sX;
#pragma unroll
    for (int t = 0; t < 8; ++t)
        xd[tid + t * 256] = xs[tid + t * 256];
    __syncthreads();

    const int wave = tid >> 5;
    const int lane = tid & 31;
    const int row  = lane & 15;                 // A-row / B-col within tile
    const int kh   = (lane >> 4) << 1;          // 0 or 2 (K half per ISA layout)
    const int tj   = tjg * 8 + wave;            // 0..63 == b index

    const float* xrow = sX + row * A_DIM;
    const float* Wcol = W + (size_t)tj * 16 + row;

    v8f acc = {};
    for (int k = 0; k < A_DIM; k += 4) {
        v2f a, b;
        // A 16x4 f32: VGPR0 = K=kh, VGPR1 = K=kh+1 (lanes 0-15: kh=0; 16-31: kh=2)
        a.x = xrow[k + kh];
        a.y = xrow[k + kh + 1];
        // B 4x16 f32: symmetric layout, N = lane&15, K = kh + vgpr
        b.x = Wcol[(size_t)(k + kh) * N_COLS];
        b.y = Wcol[(size_t)(k + kh + 1) * N_COLS];
        acc = __builtin_amdgcn_wmma_f32_16x16x4_f32(
            /*neg_a=*/false, a, /*neg_b=*/false, b,
            /*c_mod=*/(short)0, acc, /*reuse_a=*/false, /*reuse_b=*/false);
    }

    // D layout: VGPR g holds tile-row (g + 8*(lane>>4)), col = lane&15
    const int c     = lane & 15;
    const int rbase = (lane >> 4) * 8;
    float* dst = M2 + ((size_t)tj * N_BATCH) * 16 + c;
#pragma unroll
    for (int g = 0; g < 8; ++g) {
        const int i = ti * 16 + rbase + g;
        dst[(size_t)i * 16] = acc[g];
    }
}

// ---------------------------------------------------------------------------
// Phase 2: O[i,b] = sum_j exp(-sum_c |M2[b][j][c] - M2[b][i][c]|)
// grid (8, 64): blockIdx.y = b slice, blockIdx.x = 256-row i chunk.
// The b slice (2048x16 f32) is streamed through LDS in 32 KB chunks of 512
// j-rows; all lanes read the same j row (LDS broadcast, conflict-free).
// Compute-bound VALU: v_sub + v_add|abs| + v_exp per element.
// ---------------------------------------------------------------------------
__global__ __launch_bounds__(256)
void mbd_pairwise_l1exp(const float* __restrict__ M2,
                        float* __restrict__ out) {
    const int b = blockIdx.y;
    const int i = blockIdx.x * 256 + threadIdx.x;
    const float* S = M2 + (size_t)b * N_BATCH * 16;

    __shared__ float sJ[512 * 16];              // 32 KB chunk of the slice

    const float4* miP = (const float4*)(S + (size_t)i * 16);
    const float4 mi0 = miP[0], mi1 = miP[1], mi2 = miP[2], mi3 = miP[3];

    float acc = 0.f;
    for (int j0 = 0; j0 < N_BATCH; j0 += 512) {
        __syncthreads();
        const float4* src = (const float4*)(S + (size_t)j0 * 16);
        float4*       dst = (float4*)sJ;
#pragma unroll
        for (int t = 0; t < 8; ++t)
            dst[threadIdx.x + t * 256] = src[threadIdx.x + t * 256];
        __syncthreads();

        for (int j = 0; j < 512; ++j) {
            const float4* mj = (const float4*)(sJ + j * 16);
            const float4 d0 = mj[0], d1 = mj[1], d2 = mj[2], d3 = mj[3];
            float d = fabsf(d0.x - mi0.x) + fabsf(d0.y - mi0.y)
                    + fabsf(d0.z - mi0.z) + fabsf(d0.w - mi0.w)
                    + fabsf(d1.x - mi1.x) + fabsf(d1.y - mi1.y)
                    + fabsf(d1.z - mi1.z) + fabsf(d1.w - mi1.w)
                    + fabsf(d2.x - mi2.x) + fabsf(d2.y - mi2.y)
                    + fabsf(d2.z - mi2.z) + fabsf(d2.w - mi2.w)
                    + fabsf(d3.x - mi3.x) + fabsf(d3.y - mi3.y)
                    + fabsf(d3.z - mi3.z) + fabsf(d3.w - mi3.w);
            acc += __expf(-d);
        }
    }

    out[(size_t)i * B_DIM + b] = acc;
}

extern "C" void kernel_launch(void* const* d_in, const int* in_sizes, int n_in,
                              void* d_out, int out_size, void* d_ws, size_t ws_size,
                              hipStream_t stream) {
    (void)in_sizes; (void)n_in; (void)out_size; (void)ws_size;
    const float* x = (const float*)d_in[0];          // [2048, 512]
    const float* T = (const float*)d_in[1];          // [512, 64, 16] -> W [512][1024]
    float* M2  = (float*)d_ws;                       // [64][2048][16] = 8 MB scratch
    float* out = (float*)d_out;                      // [2048, 64]

    // 128 i-tiles * 8 tj-groups = 1024 blocks, 8 waves (tiles) per block
    mbd_gemm_wmma<<<dim3(1024), dim3(256), 0, stream>>>(x, T, M2);
    // 8 i-chunks * 64 b-slices
    mbd_pairwise_l1exp<<<dim3(8, 64), dim3(256), 0, stream>>>(M2, out);
}